// SelfAttention_14224931684607
// MI455X (gfx1250) — compile-verified
//
#include <hip/hip_runtime.h>
#include <stdint.h>

// ---------------------------------------------------------------------------
// Swin window attention, fp32 end-to-end (memory-bound: ~1.9 GB HBM traffic
// vs 80 GFLOP => keep exact fp32, use V_WMMA_F32_16X16X4_F32 matrix pipe).
// All LDS tiles are laid out so every WMMA fragment is a contiguous, 8-byte
// aligned pair (single ds_load_b64, no repack movs), with strides chosen for
// full 64-bank spread. Global->LDS staging uses GLOBAL_LOAD_ASYNC_TO_LDS_B128
// when the toolchain exposes it (ASYNCcnt path), else scalar fallback.
// ---------------------------------------------------------------------------

typedef float v2f __attribute__((ext_vector_type(2)));
typedef float v8f __attribute__((ext_vector_type(8)));
typedef int   v4i __attribute__((ext_vector_type(4)));

#define NTOK   64        // tokens per window
#define NH     6         // heads
#define HD     30        // head dim
#define HDP    32        // padded head dim
#define NWIN   4096      // windows
#define WNUM   1024      // mask windows

#define SA     20        // LDS stride (floats) for GEMM A  [m][k]
#define SB     20        // LDS stride for GEMM B transposed [n][k]
#define SQ     36        // LDS stride for Q/K tiles [t][d]  (16B aligned rows)
#define SV     68        // LDS stride for V transposed [d][t] and probs

#if defined(__gfx1250__) && __has_builtin(__builtin_amdgcn_global_load_async_to_lds_b128)
#define HAVE_ASYNC 1
#else
#define HAVE_ASYNC 0
#endif

static __device__ __forceinline__ v8f wmma_f32(v2f a, v2f b, v8f c) {
  // 8 args: (neg_a, A, neg_b, B, c_mod, C, reuse_a, reuse_b)
  return __builtin_amdgcn_wmma_f32_16x16x4_f32(false, a, false, b, (short)0, c,
                                               false, false);
}

#if HAVE_ASYNC
static __device__ __forceinline__ void async_cp16(const float* g, float* l) {
  // Builtin signature (from compiler diagnostic): param 1 = addrspace(1) v4i*.
  // generic->AS casts via integer round trip (LDS generic low 32 bits are the
  // LDS byte offset per the flat-aperture rules); const dropped via the cast.
  v4i __attribute__((address_space(1)))* gp =
      (v4i __attribute__((address_space(1)))*)(uintptr_t)g;
  v4i __attribute__((address_space(3)))* lp =
      (v4i __attribute__((address_space(3)))*)(uint32_t)(uintptr_t)l;
  __builtin_amdgcn_global_load_async_to_lds_b128(gp, lp, 0, 0);
}
static __device__ __forceinline__ void wait_async0() {
#if __has_builtin(__builtin_amdgcn_s_wait_asynccnt)
  __builtin_amdgcn_s_wait_asynccnt(0);
#else
  asm volatile("s_wait_asynccnt 0" ::: "memory");
#endif
}
#endif

// ---------------------------------------------------------------------------
// Generic fp32 WMMA GEMM: C[M x Ndim] = A[M x Kdim] * B[Kdim x Ndim] (+bias)
// Block: 256 threads (8 waves), tile 128(M) x 64(N); each wave owns 32x32.
// mode 0: plain row-major store (+bias), guarded on Ndim.
// mode 1: QKV scatter-store into [b][s][h][64][32] padded layout.
// ---------------------------------------------------------------------------
__global__ __launch_bounds__(256) void gemm_wmma(
    const float* __restrict__ A, const float* __restrict__ Bm,
    float* __restrict__ Cout, int Ndim, int Kdim, int mode,
    const float* __restrict__ bias) {
  __shared__ float As[128 * SA];   // [m][k]
  __shared__ float Bs[64 * SB];    // transposed: [n][k]

  const int tid  = threadIdx.x;
  const int wave = tid >> 5;
  const int lane = tid & 31;
  const int lr   = lane & 15;      // M / N position within fragment
  const int lh   = lane >> 4;      // half-wave selector (K=0/1 vs K=2/3)
  const int wm   = wave >> 1;      // 0..3
  const int wn   = wave & 1;       // 0..1
  const int m0   = blockIdx.y * 128;
  const int n0   = blockIdx.x * 64;

  v8f acc[2][2] = {};

  for (int k0 = 0; k0 < Kdim; k0 += 16) {
    const bool full = (k0 + 16 <= Kdim);
#if HAVE_ASYNC
    if (full) {
      // A block 128x16: 512 x b128 chunks, ASYNCcnt-tracked DMA into LDS.
      for (int i = tid; i < 512; i += 256) {
        const int r = i >> 2, kq = (i & 3) << 2;           // SA*r+kq % 4 == 0
        async_cp16(A + (long)(m0 + r) * Kdim + (k0 + kq), &As[r * SA + kq]);
      }
    } else
#endif
    {
      for (int i = tid; i < 128 * 16; i += 256) {
        const int m = i >> 4, k = i & 15;
        As[m * SA + k] = (k0 + k < Kdim) ? A[(long)(m0 + m) * Kdim + (k0 + k)] : 0.f;
      }
    }
    // B block 16x64, stored transposed [n][k] so fragments are contiguous.
    for (int i = tid; i < 16 * 64; i += 256) {
      const int k = i >> 6, n = i & 63;                    // coalesced on n
      float v = 0.f;
      if ((k0 + k < Kdim) && (n0 + n < Ndim)) v = Bm[(long)(k0 + k) * Ndim + (n0 + n)];
      Bs[n * SB + k] = v;
    }
#if HAVE_ASYNC
    if (full) wait_async0();
#endif
    __syncthreads();

#pragma unroll
    for (int kk = 0; kk < 16; kk += 4) {
      const int kc = kk + 2 * lh;  // A layout: lanes<16 -> K=kk,kk+1 ; lanes>=16 -> kk+2,kk+3
      v2f a[2], b[2];
#pragma unroll
      for (int mt = 0; mt < 2; mt++)
        a[mt] = *(const v2f*)&As[(wm * 32 + mt * 16 + lr) * SA + kc];
#pragma unroll
      for (int nt = 0; nt < 2; nt++)
        b[nt] = *(const v2f*)&Bs[(wn * 32 + nt * 16 + lr) * SB + kc];
#pragma unroll
      for (int mt = 0; mt < 2; mt++)
#pragma unroll
        for (int nt = 0; nt < 2; nt++)
          acc[mt][nt] = wmma_f32(a[mt], b[nt], acc[mt][nt]);
    }
    __syncthreads();
  }

  // Epilogue. C/D layout: vgpr i, lanes 0-15 -> row i, lanes 16-31 -> row i+8.
#pragma unroll
  for (int mt = 0; mt < 2; mt++)
#pragma unroll
    for (int nt = 0; nt < 2; nt++)
#pragma unroll
      for (int i = 0; i < 8; i++) {
        const int row = m0 + wm * 32 + mt * 16 + i + 8 * lh;
        const int col = n0 + wn * 32 + nt * 16 + lr;
        const float v = acc[mt][nt][i];
        if (mode == 0) {
          if (col < Ndim) Cout[(long)row * Ndim + col] = v + (bias ? bias[col] : 0.f);
        } else {
          if (col < 3 * NH * HD) {
            const int s = col / 180, r = col % 180, h = r / HD, d = r % HD;
            const int b_ = row >> 6, t = row & 63;
            Cout[(((long)(b_ * 3 + s) * NH + h) << 11) + (t << 5) + d] = v;
          }
        }
      }
}

// ---------------------------------------------------------------------------
// Per-(window, head) attention: scores -> +bias +mask -> softmax -> P*V.
// Block: 128 threads (4 waves); wave w owns score rows [16w, 16w+16).
// ---------------------------------------------------------------------------
__global__ __launch_bounds__(128) void attn_kernel(
    const float* __restrict__ qkv, const float* __restrict__ mask,
    const float* __restrict__ bias_table, float* __restrict__ Z) {
  __shared__ float Qs[NTOK * SQ];    // [t][d]
  __shared__ float Ks[NTOK * SQ];    // [t][d]
  __shared__ float Vt[HDP * SV];     // transposed: [d][t]
  __shared__ float Ps[4][16 * SV];   // per-wave 16x64 prob strip [row][t]

  const int h    = blockIdx.x;       // head
  const int b    = blockIdx.y;       // window
  const int tid  = threadIdx.x;
  const int w    = tid >> 5;
  const int lane = tid & 31;
  const int lr   = lane & 15;
  const int lh   = lane >> 4;

  const float* baseq = qkv + (((long)b * 3 + 0) * NH + h) * (NTOK * HDP);
  const float* basek = qkv + (((long)b * 3 + 1) * NH + h) * (NTOK * HDP);
  const float* basev = qkv + (((long)b * 3 + 2) * NH + h) * (NTOK * HDP);

#if HAVE_ASYNC
  for (int i = tid; i < 512; i += 128) {          // 64x32 floats, b128 chunks
    const int t = i >> 3, dq = (i & 7) << 2;      // SQ*t+dq % 4 == 0
    async_cp16(baseq + t * HDP + dq, &Qs[t * SQ + dq]);
    async_cp16(basek + t * HDP + dq, &Ks[t * SQ + dq]);
  }
  for (int i = tid; i < NTOK * HDP; i += 128) {   // V transposed (scalar)
    const int t = i >> 5, d = i & 31;
    Vt[d * SV + t] = (d < HD) ? basev[i] : 0.f;
  }
  wait_async0();
  for (int i = tid; i < NTOK; i += 128) {         // zero K-dim pads (d=30,31)
    Qs[i * SQ + 30] = 0.f; Qs[i * SQ + 31] = 0.f;
    Ks[i * SQ + 30] = 0.f; Ks[i * SQ + 31] = 0.f;
  }
#else
  for (int i = tid; i < NTOK * HDP; i += 128) {
    const int t = i >> 5, d = i & 31;
    const bool ok = d < HD;                       // zero-pad head dim 30 -> 32
    Qs[t * SQ + d] = ok ? baseq[i] : 0.f;
    Ks[t * SQ + d] = ok ? basek[i] : 0.f;
    Vt[d * SV + t] = ok ? basev[i] : 0.f;
  }
#endif
  __syncthreads();

  // ---- scores: S = Q * K^T, K-loop over padded head dim (32) ----
  v8f S[4] = {};
#pragma unroll
  for (int kk = 0; kk < HDP; kk += 4) {
    const int kc = kk + 2 * lh;
    const v2f a = *(const v2f*)&Qs[(w * 16 + lr) * SQ + kc];
#pragma unroll
    for (int j = 0; j < 4; j++) {
      const v2f bb = *(const v2f*)&Ks[(j * 16 + lr) * SQ + kc];  // B(r,n)=K[n][kc+r]
      S[j] = wmma_f32(a, bb, S[j]);
    }
  }

  // ---- scale + relative bias + mask, then rowwise softmax in-register ----
  const float scale = 0.18257418583505536f;   // 1/sqrt(30)
  const float* mrow = mask + (long)(b & (WNUM - 1)) * (NTOK * NTOK);
#pragma unroll
  for (int i = 0; i < 8; i++) {
    const int m  = w * 16 + i + 8 * lh;       // token row (0..63)
    const int mi = m >> 3, mj = m & 7;
    float val[4];
    float mx = -3.4e38f;
#pragma unroll
    for (int j = 0; j < 4; j++) {
      const int n  = j * 16 + lr;
      const int ni = n >> 3, nj = n & 7;
      const int rel = (mi - ni + 7) * 15 + (mj - nj + 7);
      float v = S[j][i] * scale + bias_table[rel * NH + h] + mrow[m * 64 + n];
      val[j] = v;
      mx = fmaxf(mx, v);
    }
    // reduce across the 16 lanes holding this row (xor<16 stays in the half)
    for (int off = 1; off < 16; off <<= 1) mx = fmaxf(mx, __shfl_xor(mx, off, 32));
    float sum = 0.f;
#pragma unroll
    for (int j = 0; j < 4; j++) { val[j] = __expf(val[j] - mx); sum += val[j]; }
    for (int off = 1; off < 16; off <<= 1) sum += __shfl_xor(sum, off, 32);
    const float inv = 1.0f / sum;
    const int prow = i + 8 * lh;
#pragma unroll
    for (int j = 0; j < 4; j++) Ps[w][prow * SV + j * 16 + lr] = val[j] * inv;
  }
  __syncthreads();

  // ---- out = P * V : M=16, N=32(padded), K=64 ----
  v8f O[2] = {};
#pragma unroll
  for (int kk = 0; kk < NTOK; kk += 4) {
    const int kc = kk + 2 * lh;
    const v2f a = *(const v2f*)&Ps[w][lr * SV + kc];
#pragma unroll
    for (int nt = 0; nt < 2; nt++) {
      const v2f bb = *(const v2f*)&Vt[(nt * 16 + lr) * SV + kc];  // B(r,n)=V[kc+r][n]
      O[nt] = wmma_f32(a, bb, O[nt]);
    }
  }

  // store z[b*64+t][h*30+d], drop padded d
#pragma unroll
  for (int nt = 0; nt < 2; nt++)
#pragma unroll
    for (int i = 0; i < 8; i++) {
      const int t = w * 16 + i + 8 * lh;
      const int d = nt * 16 + lr;
      if (d < HD) Z[((long)b * NTOK + t) * 180 + h * HD + d] = O[nt][i];
    }
}

// ---------------------------------------------------------------------------
extern "C" void kernel_launch(void* const* d_in, const int* in_sizes, int n_in,
                              void* d_out, int out_size, void* d_ws, size_t ws_size,
                              hipStream_t stream) {
  const float* windows    = (const float*)d_in[0];  // (4096, 64, 180)
  const float* mask       = (const float*)d_in[1];  // (1024, 64, 64)
  const float* w_qkv      = (const float*)d_in[2];  // (180, 540)
  const float* bias_table = (const float*)d_in[3];  // (225, 6)
  const float* w_proj     = (const float*)d_in[4];  // (180, 180)
  const float* b_proj     = (const float*)d_in[5];  // (180,)
  float* out = (float*)d_out;                       // (4096, 64, 180)

  float* qkv = (float*)d_ws;                                  // [b][s][h][64][32]
  float* Z   = qkv + (size_t)NWIN * 3 * NH * NTOK * HDP;      // (262144, 180)

  const int M = NWIN * NTOK;    // 262144 rows
  // 1) QKV projection: M x 540, K = 180 (mode 1 scatter into padded layout)
  gemm_wmma<<<dim3(9, M / 128), dim3(256), 0, stream>>>(
      windows, w_qkv, qkv, 540, 180, 1, nullptr);
  // 2) windowed attention per (window, head)
  attn_kernel<<<dim3(NH, NWIN), dim3(128), 0, stream>>>(qkv, mask, bias_table, Z);
  // 3) output projection: M x 180, K = 180, + b_proj
  gemm_wmma<<<dim3(3, M / 128), dim3(256), 0, stream>>>(
      Z, w_proj, out, 180, 180, 0, b_proj);
}